// Reduce_9783935500521
// MI455X (gfx1250) — compile-verified
//
#include <hip/hip_runtime.h>
#include <stdint.h>

// Batched masked unsorted_segment_sum for MI455X (gfx1250, wave32).
// data:        [B=16, L=16384, F=128] f32
// segment_ids: [B, L] i64  (values in [0, N))
// target:      [B, N=2048, 1] (shape only)
// data_mask:   [B, L] bool (1 byte per element, jax bool layout)
// out:         [B, N, F] f32
//
// Memory-bound scatter: ~162 MB total traffic -> ~7 us at 23.3 TB/s.
// Output (16 MB) stays resident in the 192 MB L2; atomics resolve in-L2.

typedef float v4f __attribute__((ext_vector_type(4)));

constexpr int kB = 16;
constexpr int kL = 16384;
constexpr int kF = 128;
constexpr int kN = 2048;
constexpr int kRows = kB * kL;   // 262144 rows, one wave32 per row
constexpr int kLShift = 14;      // log2(L)

// Non-returning device-scope f32 atomic add, guaranteed to lower to the
// hardware GLOBAL_ATOMIC_ADD_F32 (no clang CAS-loop expansion). Tracked by
// STOREcnt; S_ENDPGM's implicit wait-idle drains it.
__device__ __forceinline__ void global_fadd_dev(float* p, float x) {
    asm volatile("global_atomic_add_f32 %0, %1, off scope:SCOPE_DEV"
                 :: "v"(p), "v"(x)
                 : "memory");
}

// One wave32 per input row: lane i owns floats [4i, 4i+4) of the row.
// A wave's 32 * 16B = 512B load is exactly one row -> perfectly coalesced
// global_load_b128 with a non-temporal hint (streamed 128 MB must not evict
// the L2-resident 16 MB accumulator).
//
// row is wave-uniform: pin it to an SGPR via readfirstlane so the mask and
// segment-id loads become scalar (SMEM/KMcnt) instead of 32-lane broadcast
// vector loads, and the ~50% masked-row skip is a pure scalar branch.
__global__ __launch_bounds__(256) void seg_scatter_kernel(
    const v4f*       __restrict__ data,   // [kRows, kF/4]
    const long long* __restrict__ ids,    // [kRows]
    const uint8_t*   __restrict__ mask,   // [kRows]
    float*           __restrict__ out)    // [kB * kN * kF]
{
    const int gid  = blockIdx.x * 256 + threadIdx.x;
    const int lane = gid & 31;
    // Wave-uniform row index, provably uniform for the compiler.
    const int row  = __builtin_amdgcn_readfirstlane(gid >> 5);

    if (!mask[row]) return;               // scalar load + scalar branch

    const int b   = row >> kLShift;
    const int seg = (int)ids[row];        // scalar load; value < 2048

    v4f v = __builtin_nontemporal_load(&data[(size_t)row * (kF / 4) + lane]);

    float* o = out + (size_t)(b * kN + seg) * kF + (lane << 2);
    global_fadd_dev(o + 0, v[0]);
    global_fadd_dev(o + 1, v[1]);
    global_fadd_dev(o + 2, v[2]);
    global_fadd_dev(o + 3, v[3]);
}

// Zero-init the accumulator surface (128-bit stores, default RT hint so the
// zeroed lines are L2-resident when the atomics start hitting them).
__global__ __launch_bounds__(256) void seg_zero_kernel(v4f* __restrict__ out, int n4) {
    int i = blockIdx.x * 256 + threadIdx.x;
    if (i < n4) {
        v4f z = {0.0f, 0.0f, 0.0f, 0.0f};
        out[i] = z;
    }
}

extern "C" void kernel_launch(void* const* d_in, const int* in_sizes, int n_in,
                              void* d_out, int out_size, void* d_ws, size_t ws_size,
                              hipStream_t stream) {
    const v4f*       data = (const v4f*)d_in[0];
    const long long* ids  = (const long long*)d_in[1];
    // d_in[2] = target: shape-only, unused.
    const uint8_t*   mask = (const uint8_t*)d_in[3];
    float*           out  = (float*)d_out;

    // 1) Zero the B*N*F accumulator (out_size = 4,194,304 f32 = 1,048,576 v4f).
    const int n4 = out_size / 4;
    seg_zero_kernel<<<(n4 + 255) / 256, 256, 0, stream>>>((v4f*)d_out, n4);

    // 2) Scatter-add: one wave32 per row -> kRows * 32 threads total.
    const int total_threads = kRows * 32;            // 8,388,608
    seg_scatter_kernel<<<total_threads / 256, 256, 0, stream>>>(data, ids, mask, out);
}